// ReasonerHead_32109175505346
// MI455X (gfx1250) — compile-verified
//
#include <hip/hip_runtime.h>
#include <math.h>

// CDNA5 / gfx1250: fp32 WMMA 16x16x4 (wave32) + double-buffered TDM panel staging.
typedef __attribute__((ext_vector_type(2))) float v2f;
typedef __attribute__((ext_vector_type(8))) float v8f;
typedef __attribute__((ext_vector_type(4))) unsigned int u32x4;
typedef __attribute__((ext_vector_type(8))) int i32x8;
typedef __attribute__((ext_vector_type(4))) int i32x4;

#define EMB 32
#define KPANELS 34          // 1088 / 32 K-panels
#define GSTRIDE 34          // dwords per 16-lane fragment group (32 data + 2 pad)
#define PANELD(NT) ((NT) * 16 * GSTRIDE)

__device__ inline v8f wmma4(v2f a, v2f b, v8f c) {
  return __builtin_amdgcn_wmma_f32_16x16x4_f32(
      false, a, false, b, (short)0, c, false, false);
}

// ---- TDM: DMA one pre-swizzled weight panel (panelD dwords) into LDS -------
__device__ inline void tdm_load_panel(const float* gsrc, unsigned ldsOff, int panelD) {
  unsigned long long ga = (unsigned long long)(uintptr_t)gsrc;
  u32x4 g0;
  g0.x = 1u;                                          // count=1, user mode
  g0.y = ldsOff;                                      // lds_addr [63:32]
  g0.z = (unsigned)(ga & 0xFFFFFFFFu);                // global_addr low
  g0.w = (unsigned)((ga >> 32) & 0x1FFFFFFu) | (2u << 30);  // addr hi | type=2
  i32x8 g1;
  g1[0] = 0x20000;                                    // data_size=4B, no mask
  g1[1] = (int)(((unsigned)panelD & 0xFFFFu) << 16);  // tensor_dim0[15:0]
  g1[2] = (int)((((unsigned)panelD >> 16) & 0xFFFFu) | (1u << 16)); // dim0 hi | dim1=1
  g1[3] = (int)((unsigned)panelD << 16);              // tile_dim0 [127:112]
  g1[4] = 1;                                          // tile_dim1 = 1
  g1[5] = panelD;                                     // tensor_dim0_stride lo
  g1[6] = 0;
  g1[7] = 0;
  i32x4 z4 = {0, 0, 0, 0};
  i32x8 z8 = {0, 0, 0, 0, 0, 0, 0, 0};
  __builtin_amdgcn_tensor_load_to_lds(g0, g1, z4, z4, z8, 0);
}

// ---- one weight panel vs one A source -------------------------------------
template <int NT, typename F>
__device__ __attribute__((always_inline)) inline
void run_panel(const v2f* Bp, F makeA, v8f* acc, int h16, int m16) {
#pragma unroll
  for (int ks = 0; ks < 8; ++ks) {
    v2f a = makeA(ks);
#pragma unroll
    for (int nt = 0; nt < NT; ++nt) {
      v2f b = Bp[((ks * NT + nt) * 2 + h16) * (GSTRIDE / 2) + m16];
      acc[nt] = wmma4(a, b, acc[nt]);
    }
  }
}

// ---- one weight panel vs two A sources (left/right share B-fragments) -----
template <typename F0, typename F1>
__device__ __attribute__((always_inline)) inline
void run_panel_dual(const v2f* Bp, F0 a0f, F1 a1f,
                    v8f* accL, v8f* accR, int h16, int m16) {
#pragma unroll
  for (int ks = 0; ks < 8; ++ks) {
    v2f a0 = a0f(ks), a1 = a1f(ks);
    v2f b0 = Bp[((ks * 2 + 0) * 2 + h16) * (GSTRIDE / 2) + m16];
    v2f b1 = Bp[((ks * 2 + 1) * 2 + h16) * (GSTRIDE / 2) + m16];
    accL[0] = wmma4(a0, b0, accL[0]);
    accL[1] = wmma4(a0, b1, accL[1]);
    accR[0] = wmma4(a1, b0, accR[0]);
    accR[1] = wmma4(a1, b1, accR[1]);
  }
}

// ---- h-stage GEMM: F(16x1088)=[P|Q|outer] @ Ws1 (NT=8), double-buffered TDM
__device__ inline void gemm_h(const float* __restrict__ WgSw,
                              float* __restrict__ sWbase, unsigned off0,
                              const float* Prow, const float* Qrow,
                              int w, int m16, int h16, v8f* acc) {
  constexpr int PD = PANELD(8);
  const unsigned off1 = off0 + PD * 4;
  const v2f* buf0 = (const v2f*)sWbase;
  const v2f* buf1 = (const v2f*)(sWbase + PD);
  const v2f* Pv = (const v2f*)Prow;
  const v2f* Qv = (const v2f*)Qrow;

  if (w == 0) { tdm_load_panel(WgSw, off0, PD); __builtin_amdgcn_s_wait_tensorcnt(0); }
  __syncthreads();
  // kp = 0 : P segment
  if (w == 0) tdm_load_panel(WgSw + (size_t)PD, off1, PD);
  run_panel<8>(buf0, [&](int ks) { return Pv[2 * ks + h16]; }, acc, h16, m16);
  if (w == 0) __builtin_amdgcn_s_wait_tensorcnt(0);
  __syncthreads();
  // kp = 1 : Q segment
  if (w == 0) tdm_load_panel(WgSw + 2 * (size_t)PD, off0, PD);
  run_panel<8>(buf1, [&](int ks) { return Qv[2 * ks + h16]; }, acc, h16, m16);
  if (w == 0) __builtin_amdgcn_s_wait_tensorcnt(0);
  __syncthreads();
  // kp = 2..33 : outer segment (hot loop, branch-free)
  for (int kp = 2; kp < KPANELS; ++kp) {
    if (w == 0 && kp + 1 < KPANELS)
      tdm_load_panel(WgSw + (size_t)(kp + 1) * PD, ((kp + 1) & 1) ? off1 : off0, PD);
    const float pv = Prow[kp - 2];
    run_panel<8>((kp & 1) ? buf1 : buf0,
                 [&](int ks) { v2f q = Qv[2 * ks + h16];
                               v2f r; r.x = pv * q.x; r.y = pv * q.y; return r; },
                 acc, h16, m16);
    if (w == 0) __builtin_amdgcn_s_wait_tensorcnt(0);
    __syncthreads();
  }
}

// ---- fused left/right GEMM vs Wa (NT=2): B-fragments shared ---------------
__device__ inline void gemm_lr(const float* __restrict__ WgSw,
                               float* __restrict__ sWbase, unsigned off0,
                               const float* P0, const float* Q0,   // na, b
                               const float* P1, const float* Q1,   // c, d
                               int w, int m16, int h16, v8f* accL, v8f* accR) {
  constexpr int PD = PANELD(2);
  const unsigned off1 = off0 + PD * 4;
  const v2f* buf0 = (const v2f*)sWbase;
  const v2f* buf1 = (const v2f*)(sWbase + PD);
  const v2f* P0v = (const v2f*)P0;  const v2f* Q0v = (const v2f*)Q0;
  const v2f* P1v = (const v2f*)P1;  const v2f* Q1v = (const v2f*)Q1;

  if (w == 0) { tdm_load_panel(WgSw, off0, PD); __builtin_amdgcn_s_wait_tensorcnt(0); }
  __syncthreads();
  // kp = 0 : P segment
  if (w == 0) tdm_load_panel(WgSw + (size_t)PD, off1, PD);
  run_panel_dual(buf0, [&](int ks) { return P0v[2 * ks + h16]; },
                       [&](int ks) { return P1v[2 * ks + h16]; },
                 accL, accR, h16, m16);
  if (w == 0) __builtin_amdgcn_s_wait_tensorcnt(0);
  __syncthreads();
  // kp = 1 : Q segment
  if (w == 0) tdm_load_panel(WgSw + 2 * (size_t)PD, off0, PD);
  run_panel_dual(buf1, [&](int ks) { return Q0v[2 * ks + h16]; },
                       [&](int ks) { return Q1v[2 * ks + h16]; },
                 accL, accR, h16, m16);
  if (w == 0) __builtin_amdgcn_s_wait_tensorcnt(0);
  __syncthreads();
  // kp = 2..33 : outer segment
  for (int kp = 2; kp < KPANELS; ++kp) {
    if (w == 0 && kp + 1 < KPANELS)
      tdm_load_panel(WgSw + (size_t)(kp + 1) * PD, ((kp + 1) & 1) ? off1 : off0, PD);
    const float pv0 = P0[kp - 2];
    const float pv1 = P1[kp - 2];
    run_panel_dual((kp & 1) ? buf1 : buf0,
                   [&](int ks) { v2f q = Q0v[2 * ks + h16];
                                 v2f r; r.x = pv0 * q.x; r.y = pv0 * q.y; return r; },
                   [&](int ks) { v2f q = Q1v[2 * ks + h16];
                                 v2f r; r.x = pv1 * q.x; r.y = pv1 * q.y; return r; },
                   accL, accR, h16, m16);
    if (w == 0) __builtin_amdgcn_s_wait_tensorcnt(0);
    __syncthreads();
  }
}

__global__ __launch_bounds__(256, 1) void ReasonerHead_32109175505346_kernel(
    const float* __restrict__ concepts,
    const float* __restrict__ bn, const float* __restrict__ ba,
    const float* __restrict__ bs1, const float* __restrict__ Ws2,
    const float* __restrict__ bs2, const int* __restrict__ idx,
    const float* __restrict__ WnS, const float* __restrict__ WaS,
    const float* __restrict__ Ws1S,   // pre-swizzled panels in d_ws
    float* __restrict__ out) {
  __shared__ __align__(16) float sW[2][PANELD(8)];    // double-buffered panel (34.8 KB)
  __shared__ __align__(16) float sT0[8][16][GSTRIDE]; // a -> na -> left
  __shared__ __align__(16) float sT1[8][16][GSTRIDE]; // b       -> right
  __shared__ __align__(16) float sT2[8][16][GSTRIDE]; // c
  __shared__ __align__(16) float sT3[8][16][GSTRIDE]; // d
  __shared__ int sIdx[8][16][4];

  const int tid  = threadIdx.x;
  const int w    = tid >> 5;
  const int lane = tid & 31;
  const int m16  = lane & 15;
  const int h16  = lane >> 4;
  const int rowBase = blockIdx.x * 128 + w * 16;
  const unsigned sWoff = (unsigned)(uintptr_t)(void*)&sW[0][0];

  const v8f vzero = {0.f, 0.f, 0.f, 0.f, 0.f, 0.f, 0.f, 0.f};
  float* T0 = &sT0[w][m16][0];
  float* T1 = &sT1[w][m16][0];
  float* T2 = &sT2[w][m16][0];
  float* T3 = &sT3[w][m16][0];

  // ---- stage indices, gather all four operand tiles ------------------------
  if (lane < 16) {
#pragma unroll
    for (int j = 0; j < 4; ++j)
      sIdx[w][lane][j] = idx[(rowBase + lane) * 4 + j];
  }
  __syncthreads();
#pragma unroll
  for (int t = 0; t < 16; ++t) {
    sT0[w][t][lane] = concepts[(size_t)sIdx[w][t][0] * EMB + lane];  // a
    sT1[w][t][lane] = concepts[(size_t)sIdx[w][t][1] * EMB + lane];  // b
    sT2[w][t][lane] = concepts[(size_t)sIdx[w][t][2] * EMB + lane];  // c
    sT3[w][t][lane] = concepts[(size_t)sIdx[w][t][3] * EMB + lane];  // d
  }

  // ==== na = a @ Wn + bn ====================================================
  if (w == 0) { tdm_load_panel(WnS, sWoff, PANELD(2)); __builtin_amdgcn_s_wait_tensorcnt(0); }
  __syncthreads();
  v8f accN[2] = {vzero, vzero};
  {
    const v2f* Av = (const v2f*)T0;
    run_panel<2>((const v2f*)&sW[0][0],
                 [&](int ks) { return Av[2 * ks + h16]; }, accN, h16, m16);
  }
  __syncthreads();   // all waves done with sW before gemm_lr re-stages it
#pragma unroll
  for (int nt = 0; nt < 2; ++nt) {
    const float bv = bn[nt * 16 + m16];
#pragma unroll
    for (int v = 0; v < 8; ++v)
      sT0[w][v + 8 * h16][nt * 16 + m16] = accN[nt][v] + bv;   // na over a
  }

  // ==== left = [na,b,outer] @ Wa ; right = [c,d,outer] @ Wa (fused) =========
  v8f accL[2] = {vzero, vzero};
  v8f accR[2] = {vzero, vzero};
  gemm_lr(WaS, &sW[0][0], sWoff, T0, T1, T2, T3, w, m16, h16, accL, accR);

  // add ba; park left in sT0, right in sT1 for the h-stage
#pragma unroll
  for (int nt = 0; nt < 2; ++nt) {
    const float bv = ba[nt * 16 + m16];
#pragma unroll
    for (int v = 0; v < 8; ++v) {
      sT0[w][v + 8 * h16][nt * 16 + m16] = accL[nt][v] + bv;
      sT1[w][v + 8 * h16][nt * 16 + m16] = accR[nt][v] + bv;
    }
  }

  // ==== h = elu([left,right,outer] @ Ws1 + bs1) =============================
  v8f accH[8] = {vzero, vzero, vzero, vzero, vzero, vzero, vzero, vzero};
  gemm_h(Ws1S, &sW[0][0], sWoff, T0, T1, w, m16, h16, accH);

  // ==== out = h @ Ws2 + bs2 (ELU fused, butterfly reduce over 16 lanes) =====
  float result[8];
#pragma unroll
  for (int v = 0; v < 8; ++v) result[v] = 0.f;
#pragma unroll
  for (int nt = 0; nt < 8; ++nt) {
    const float b1 = bs1[nt * 16 + m16];
    const float w2 = Ws2[nt * 16 + m16];
#pragma unroll
    for (int v = 0; v < 8; ++v) {
      float x = accH[nt][v] + b1;
      x = (x > 0.f) ? x : (expf(x) - 1.f);           // ELU (alpha=1)
      result[v] += x * w2;
    }
  }
  const float b2 = bs2[0];
#pragma unroll
  for (int v = 0; v < 8; ++v) {
    float r = result[v];
    r += __shfl_xor(r, 1, 32);
    r += __shfl_xor(r, 2, 32);
    r += __shfl_xor(r, 4, 32);
    r += __shfl_xor(r, 8, 32);
    if (m16 == 0) out[rowBase + v + 8 * h16] = r + b2;
  }
}

// ---- prep: rewrite row-major W (32*KP x NT*16) into WMMA B-fragment panels -
__global__ void prep_swizzle_kernel(const float* __restrict__ W,
                                    float* __restrict__ dst, int KP, int NT) {
  const int NC = NT * 16;
  const int panelD = NT * 16 * GSTRIDE;
  const long total = (long)KP * panelD;
  for (long d = (long)blockIdx.x * blockDim.x + threadIdx.x; d < total;
       d += (long)gridDim.x * blockDim.x) {
    const int kp = (int)(d / panelD);
    const int t  = (int)(d % panelD);
    const int g = t / GSTRIDE, off = t % GSTRIDE;
    float v = 0.f;
    if (off < 32) {
      const int n = off >> 1, r = off & 1, h = g & 1, gg = g >> 1;
      const int nt = gg % NT, ks = gg / NT;
      const int row = kp * 32 + 4 * ks + 2 * h + r;
      const int col = nt * 16 + n;
      v = W[row * NC + col];
    }
    dst[d] = v;
  }
}

extern "C" void kernel_launch(void* const* d_in, const int* in_sizes, int n_in,
                              void* d_out, int out_size, void* d_ws, size_t ws_size,
                              hipStream_t stream) {
  const float* concepts = (const float*)d_in[0];
  const float* Wn  = (const float*)d_in[1];
  const float* bn  = (const float*)d_in[2];
  const float* Wa  = (const float*)d_in[3];
  const float* ba  = (const float*)d_in[4];
  const float* Ws1 = (const float*)d_in[5];
  const float* bs1 = (const float*)d_in[6];
  const float* Ws2 = (const float*)d_in[7];
  const float* bs2 = (const float*)d_in[8];
  const int*   idx = (const int*)d_in[9];
  float* out = (float*)d_out;

  // d_ws layout (floats): WnS[1 panel NT=2] | WaS[34 panels NT=2] | Ws1S[34 panels NT=8]
  float* WnS  = (float*)d_ws;
  float* WaS  = WnS + PANELD(2);
  float* Ws1S = WaS + (size_t)KPANELS * PANELD(2);

  prep_swizzle_kernel<<<8,   256, 0, stream>>>(Wn,  WnS,  1,       2);
  prep_swizzle_kernel<<<64,  256, 0, stream>>>(Wa,  WaS,  KPANELS, 2);
  prep_swizzle_kernel<<<256, 256, 0, stream>>>(Ws1, Ws1S, KPANELS, 8);

  const int N = in_sizes[9] / 4;           // rows (idx is N x 4)
  const int blocks = N / 128;              // 8 waves * 16 rows per block
  ReasonerHead_32109175505346_kernel<<<blocks, 256, 0, stream>>>(
      concepts, bn, ba, bs1, Ws2, bs2, idx, WnS, WaS, Ws1S, out);
}